// Voxelizer_54279796687046
// MI455X (gfx1250) — compile-verified
//
#include <hip/hip_runtime.h>

#define NV  262144         // num_voxels (static in the reference)
#define NBC 64             // B*C feature planes
#define CH  8              // planes per load/consume chunk (8 x float4 = 32 VGPRs in flight)

typedef float v4f __attribute__((ext_vector_type(4)));
typedef int   v4i __attribute__((ext_vector_type(4)));

// ---------------- vectorized path (npts % 4 == 0): 4 points per thread ------

__global__ __launch_bounds__(256) void vox_scatter4(
    const float* __restrict__ feat, const int* __restrict__ idx,
    float* __restrict__ sums, float* __restrict__ counts, int npts)
{
    int t  = blockIdx.x * 256 + threadIdx.x;
    int n0 = t * 4;
    if (n0 >= npts) return;

    v4i vi = *(const v4i*)(idx + n0);            // 16B-aligned: n0 % 4 == 0
    unsafeAtomicAdd(counts + vi.x, 1.0f);
    unsafeAtomicAdd(counts + vi.y, 1.0f);
    unsafeAtomicAdd(counts + vi.z, 1.0f);
    unsafeAtomicAdd(counts + vi.w, 1.0f);

    for (int base = 0; base < NBC; base += CH) {
        v4f x[CH];
        // phase 1: fire CH b128 NT loads (kept in flight together)
        #pragma unroll
        for (int j = 0; j < CH; ++j)
            x[j] = __builtin_nontemporal_load(
                       (const v4f*)(feat + (size_t)(base + j) * npts + n0));
        // phase 2: 4*CH no-return fp32 atomics (STOREcnt, fire-and-forget)
        #pragma unroll
        for (int j = 0; j < CH; ++j) {
            float* s = sums + (size_t)(base + j) * NV;
            unsafeAtomicAdd(s + vi.x, x[j].x);
            unsafeAtomicAdd(s + vi.y, x[j].y);
            unsafeAtomicAdd(s + vi.z, x[j].z);
            unsafeAtomicAdd(s + vi.w, x[j].w);
        }
    }
}

__global__ __launch_bounds__(256) void vox_gather4(
    const float* __restrict__ sums, const float* __restrict__ inv,
    const int* __restrict__ idx, float* __restrict__ out, int npts)
{
    int t  = blockIdx.x * 256 + threadIdx.x;
    int n0 = t * 4;
    if (n0 >= npts) return;

    v4i vi = *(const v4i*)(idx + n0);
    v4f w;
    w.x = inv[vi.x];
    w.y = inv[vi.y];
    w.z = inv[vi.z];
    w.w = inv[vi.w];

    for (int base = 0; base < NBC; base += CH) {
        v4f r[CH];
        // phase 1: 4*CH random gather loads in flight (L2-resident sums)
        #pragma unroll
        for (int j = 0; j < CH; ++j) {
            const float* s = sums + (size_t)(base + j) * NV;
            r[j].x = s[vi.x];
            r[j].y = s[vi.y];
            r[j].z = s[vi.z];
            r[j].w = s[vi.w];
        }
        // phase 2: scale and stream out with b128 NT stores
        #pragma unroll
        for (int j = 0; j < CH; ++j)
            __builtin_nontemporal_store(r[j] * w,
                (v4f*)(out + (size_t)(base + j) * npts + n0));
    }
}

// ---------------- scalar fallback (any npts) --------------------------------

__global__ __launch_bounds__(256) void vox_scatter1(
    const float* __restrict__ feat, const int* __restrict__ idx,
    float* __restrict__ sums, float* __restrict__ counts, int npts)
{
    int n = blockIdx.x * 256 + threadIdx.x;
    if (n >= npts) return;
    int v = idx[n];
    unsafeAtomicAdd(counts + v, 1.0f);
    for (int base = 0; base < NBC; base += CH) {
        float x[CH];
        #pragma unroll
        for (int j = 0; j < CH; ++j)
            x[j] = __builtin_nontemporal_load(feat + (size_t)(base + j) * npts + n);
        #pragma unroll
        for (int j = 0; j < CH; ++j)
            unsafeAtomicAdd(sums + (size_t)(base + j) * NV + v, x[j]);
    }
}

__global__ __launch_bounds__(256) void vox_gather1(
    const float* __restrict__ sums, const float* __restrict__ inv,
    const int* __restrict__ idx, float* __restrict__ out, int npts)
{
    int n = blockIdx.x * 256 + threadIdx.x;
    if (n >= npts) return;
    int v = idx[n];
    float w = inv[v];
    for (int base = 0; base < NBC; base += CH) {
        float r[CH];
        #pragma unroll
        for (int j = 0; j < CH; ++j)
            r[j] = sums[(size_t)(base + j) * NV + v];
        #pragma unroll
        for (int j = 0; j < CH; ++j)
            __builtin_nontemporal_store(r[j] * w, out + (size_t)(base + j) * npts + n);
    }
}

// ---------------- per-voxel reciprocal --------------------------------------

__global__ __launch_bounds__(256) void vox_inv(float* __restrict__ counts)
{
    int v = blockIdx.x * 256 + threadIdx.x;
    float c = counts[v];
    counts[v] = 1.0f / fmaxf(c, 1.0f);
}

// ---------------- launcher ---------------------------------------------------

extern "C" void kernel_launch(void* const* d_in, const int* in_sizes, int n_in,
                              void* d_out, int out_size, void* d_ws, size_t ws_size,
                              hipStream_t stream)
{
    const float* feat = (const float*)d_in[0];
    const int*   idx  = (const int*)d_in[1];
    const int npts = in_sizes[1];          // N = 1,000,000

    float* sums   = (float*)d_ws;                    // NBC*NV floats = 64 MB (L2-resident)
    float* counts = sums + (size_t)NBC * NV;         // NV floats = 1 MB
    size_t zero_bytes = ((size_t)NBC * NV + (size_t)NV) * sizeof(float);

    hipMemsetAsync(d_ws, 0, zero_bytes, stream);

    if ((npts & 3) == 0) {
        int nthreads = npts / 4;
        int blocks = (nthreads + 255) / 256;
        vox_scatter4<<<blocks, 256, 0, stream>>>(feat, idx, sums, counts, npts);
        vox_inv<<<NV / 256, 256, 0, stream>>>(counts);
        vox_gather4<<<blocks, 256, 0, stream>>>(sums, counts, idx, (float*)d_out, npts);
    } else {
        int blocks = (npts + 255) / 256;
        vox_scatter1<<<blocks, 256, 0, stream>>>(feat, idx, sums, counts, npts);
        vox_inv<<<NV / 256, 256, 0, stream>>>(counts);
        vox_gather1<<<blocks, 256, 0, stream>>>(sums, counts, idx, (float*)d_out, npts);
    }
}